// Conv3DSubmModule_82755429859556
// MI455X (gfx1250) — compile-verified
//
#include <hip/hip_runtime.h>

typedef __attribute__((ext_vector_type(2))) float v2f;
typedef __attribute__((ext_vector_type(8))) float v8f;

#define KVOL 27
#define DIN  64
#define DOUT 64
#define WPF  (DIN * DOUT)   // floats per W[k] slice = 4096

// One wave computes a 16-row x 64-col output tile with fp32 WMMA
// (V_WMMA_F32_16X16X4_F32: A=16x4, B=4x16, C/D=16x16, all fp32).
// 8 waves/block share W[k] staged in double-buffered LDS (2 x 32 KB declared, 16 KB of each used).
//
// LDS layout (pair-interleaved + bank swizzle):
//   element (d, col) of W[k]  ->  w_lds[buf][ (dp*64 + pcol)*2 + (d&1) ]
//   where dp = d>>1, pcol = (col + 16*(dp&1)) & 63.
// A B-fragment (W[d][col], W[d+1][col]) is then one contiguous ds_load_b64,
// and the swizzle puts lane halves (hi == dp&1 at read time) on disjoint banks.
__global__ __launch_bounds__(256) void subm_conv3d_wmma_f32(
    const float* __restrict__ feats,          // [N, 64]
    const long long* __restrict__ indices,    // [27, N]
    const float* __restrict__ weight,         // [27, 64, 64]
    float* __restrict__ out,                  // [N, 64]
    int n_out)
{
    __shared__ float w_lds[2][WPF * 2 / 2 * 2];   // 2 x 8192 floats = 64 KB

    const int tid  = threadIdx.x;
    const int wave = tid >> 5;                // 0..7
    const int lane = tid & 31;
    const int r    = lane & 15;               // row-within-tile / B column
    const bool hi  = lane >= 16;              // upper half-wave

    const int n0 = (blockIdx.x * 8 + wave) * 16;   // first output row of this wave

    // ---- staging geometry: thread handles rows (2*dp, 2*dp+1), cols colBase..colBase+7
    const int dp       = tid >> 3;                         // 0..31
    const int colBase  = (tid & 7) * 8;                    // 0,8,...,56
    const int physBase = (colBase + ((dp & 1) ? 16 : 0)) & 63;

    auto stage = [&](int k, int buf) {
        const float* row0 = weight + (size_t)k * WPF + (2 * dp) * DOUT + colBase;
        const float4 r0a = *(const float4*)(row0 + 0);
        const float4 r0b = *(const float4*)(row0 + 4);
        const float4 r1a = *(const float4*)(row0 + DOUT + 0);
        const float4 r1b = *(const float4*)(row0 + DOUT + 4);
        float4* dst = (float4*)&w_lds[buf][(dp * 64 + physBase) * 2];
        dst[0] = make_float4(r0a.x, r1a.x, r0a.y, r1a.y);
        dst[1] = make_float4(r0a.z, r1a.z, r0a.w, r1a.w);
        dst[2] = make_float4(r0b.x, r1b.x, r0b.y, r1b.y);
        dst[3] = make_float4(r0b.z, r1b.z, r0b.w, r1b.w);
    };

    v8f acc0 = {}, acc1 = {}, acc2 = {}, acc3 = {};

    // prologue: stage W[0] into buffer 0
    stage(0, 0);
    __syncthreads();

    int buf = 0;
    for (int k = 0; k < KVOL; ++k) {
        // Prefetch W[k+1] into the other buffer. Safe with a single barrier:
        // the other buffer was last *read* during iteration k-1, whose
        // end-of-iteration barrier has already completed.
        if (k + 1 < KVOL)
            stage(k + 1, buf ^ 1);

        // ---- gather neighbor row for this kernel offset (A fragments)
        long long idx = -1;
        if (n0 + r < n_out)
            idx = __builtin_nontemporal_load(&indices[(size_t)k * n_out + (n0 + r)]);
        const bool valid = idx >= 0;
        const float* arow = feats + (size_t)(valid ? idx : 0) * DIN + (hi ? 2 : 0);

        // A layout for 16x4 fp32 tile, d-chunk c (d = 4c..4c+3):
        //   lane<16 : row r, holds d=4c+0 (v0), 4c+1 (v1)
        //   lane>=16: row r, holds d=4c+2 (v0), 4c+3 (v1)
        v2f afrag[16];
#pragma unroll
        for (int c = 0; c < 16; ++c)
            afrag[c] = *(const v2f*)(arow + 4 * c);
#pragma unroll
        for (int c = 0; c < 16; ++c) {
            if (!valid) { afrag[c].x = 0.0f; afrag[c].y = 0.0f; }
        }

        // ---- 16 d-chunks x 4 column tiles of WMMA
        const float* wb = &w_lds[buf][0];
#pragma unroll
        for (int j = 0; j < 4; ++j) {
            // swizzled physical column for this lane (dp&1 == hi at read time)
            const int pcol = ((16 * j + r) + (hi ? 16 : 0)) & 63;
            // base dword index for c=0: dp_read = 2c+hi
            const float* bbase = wb + ((hi ? 64 : 0) + pcol) * 2;
            v8f& acc = (j == 0) ? acc0 : (j == 1) ? acc1 : (j == 2) ? acc2 : acc3;
#pragma unroll
            for (int c = 0; c < 16; ++c) {
                // one contiguous b64: (W[4c+2hi][col], W[4c+2hi+1][col])
                const v2f bfrag = *(const v2f*)(bbase + c * (128 * 2));
                acc = __builtin_amdgcn_wmma_f32_16x16x4_f32(
                        /*neg_a=*/false, afrag[c],
                        /*neg_b=*/false, bfrag,
                        /*c_mod=*/(short)0, acc,
                        /*reuse_a=*/false, /*reuse_b=*/false);
            }
        }

        __syncthreads();     // all reads of buf done + all writes of buf^1 visible
        buf ^= 1;
    }

    // ---- store D: VGPR v of lane l holds (row = v + 8*hi, col = r) of the 16x16 tile
    // Non-temporal: keep feats resident in L2 (the gather working set).
#pragma unroll
    for (int j = 0; j < 4; ++j) {
        const v8f& acc = (j == 0) ? acc0 : (j == 1) ? acc1 : (j == 2) ? acc2 : acc3;
#pragma unroll
        for (int v = 0; v < 8; ++v) {
            const int row = n0 + v + (hi ? 8 : 0);
            if (row < n_out)
                __builtin_nontemporal_store(acc[v], &out[(size_t)row * DOUT + j * 16 + r]);
        }
    }
}

extern "C" void kernel_launch(void* const* d_in, const int* in_sizes, int n_in,
                              void* d_out, int out_size, void* d_ws, size_t ws_size,
                              hipStream_t stream) {
    const float*     feats   = (const float*)d_in[0];      // [N, 64] f32
    const long long* indices = (const long long*)d_in[1];  // [27, N] i64
    const float*     weight  = (const float*)d_in[2];      // [27, 64, 64] f32
    float*           out     = (float*)d_out;              // [N, 64] f32

    const int n = in_sizes[0] / DIN;                       // N = 400000
    const int blocks = (n + 127) / 128;                    // 128 rows per block (8 waves x 16)
    subm_conv3d_wmma_f32<<<blocks, 256, 0, stream>>>(feats, indices, weight, out, n);
}